// Conv1dMultiscaleLocalization_67602785239353
// MI455X (gfx1250) — compile-verified
//
#include <hip/hip_runtime.h>
#include <hip/hip_bf16.h>

typedef float v2f __attribute__((ext_vector_type(2)));
typedef float v8f __attribute__((ext_vector_type(8)));

#define BATCH 16
#define HH 768
#define WW 768
#define HW (HH * WW)
#define TOPK 4096
#define CAP 65536
#define NEGF -1.0e30f

// ---------------- Kernel A: multiscale step-conv via fp32 WMMA ----------------
// Tile: 64 (x) x 32 (y) outputs per workgroup, 256 threads = 8 waves.
#define TX 64
#define TY 32
#define KP 68   // 65 taps padded to multiple of 4

__constant__ int c_ks[7] = {3, 9, 15, 21, 31, 51, 65};

__device__ __forceinline__ float kweight(int n, int t) {
    if (n >= 7 || t > 64) return 0.0f;
    int w = c_ks[n];
    int half = (w - 1) / 2;
    int d = t - 32;
    if (d == 0 || d < -half || d > half) return 0.0f;
    return (d < 0 ? 1.0f : -1.0f) / (float)(w - 1);
}

__global__ __launch_bounds__(256)
void convA_kernel(const float* __restrict__ C, const float* __restrict__ S,
                  float* __restrict__ resp) {
    __shared__ float Slds[TY][TX + 64 + 4];     // cols 0..131 -> gx = x0-32+i
    __shared__ float Clds[TY + 68][TX];         // rows 0..99  -> gy = y0-32+r
    __shared__ float Bw[KP][16];

    const int x0 = blockIdx.x * TX;
    const int y0 = blockIdx.y * TY;
    const int b  = blockIdx.z;
    const long base = (long)b * HW;
    const int tid = threadIdx.x;

    // kernel-weight matrix (constant, tiny)
    for (int e = tid; e < KP * 16; e += 256) {
        int t = e >> 4, n = e & 15;
        Bw[t][n] = kweight(n, t);
    }
    // S tile: 32 rows x 132 cols
    for (int e = tid; e < TY * 132; e += 256) {
        int r = e / 132, c = e % 132;
        int gx = x0 - 32 + c;
        float v = 0.0f;
        if (gx >= 0 && gx < WW) v = S[base + (long)(y0 + r) * WW + gx];
        Slds[r][c] = v;
    }
    // C tile: 100 rows x 64 cols
    for (int e = tid; e < 100 * TX; e += 256) {
        int r = e / TX, c = e % TX;
        int gy = y0 - 32 + r;
        float v = 0.0f;
        if (gy >= 0 && gy < HH) v = C[base + (long)gy * WW + x0 + c];
        Clds[r][c] = v;
    }
    __syncthreads();

    const int lane = tid & 31;
    const int wid  = tid >> 5;
    const int m    = lane & 15;        // M index (pixel within 16-group / N index for B)
    const int koff = (lane >> 4) * 2;  // K sub-offset per A/B fragment layout

    // 128 chains of 16 pixels; 8 waves
    for (int q = wid; q < TY * (TX / 16); q += 8) {
        const int ty = q >> 2;
        const int cx = (q & 3) * 16;

        v8f acc = {0.f, 0.f, 0.f, 0.f, 0.f, 0.f, 0.f, 0.f};

        // horizontal conv on S: A[m][t] = Slds[ty][cx+m+t]
        for (int kc = 0; kc < KP; kc += 4) {
            v2f a, bf;
            a.x  = Slds[ty][cx + m + kc + koff];
            a.y  = Slds[ty][cx + m + kc + koff + 1];
            bf.x = Bw[kc + koff][m];
            bf.y = Bw[kc + koff + 1][m];
            acc = __builtin_amdgcn_wmma_f32_16x16x4_f32(false, a, false, bf,
                                                        (short)0, acc, false, false);
        }
        // vertical conv on C: A[m][t] = Clds[ty+t][cx+m]
        for (int kc = 0; kc < KP; kc += 4) {
            v2f a, bf;
            a.x  = Clds[ty + kc + koff][cx + m];
            a.y  = Clds[ty + kc + koff + 1][cx + m];
            bf.x = Bw[kc + koff][m];
            bf.y = Bw[kc + koff + 1][m];
            acc = __builtin_amdgcn_wmma_f32_16x16x4_f32(false, a, false, bf,
                                                        (short)0, acc, false, false);
        }

        // max over scales n=0..6 (columns of D); butterfly within 8-lane groups
        const long rowbase = base + (long)(y0 + ty) * WW + x0 + cx;
        #pragma unroll
        for (int i = 0; i < 8; ++i) {
            float v = acc[i];
            if ((lane & 15) >= 7) v = NEGF;
            v = fmaxf(v, __shfl_xor(v, 1, 32));
            v = fmaxf(v, __shfl_xor(v, 2, 32));
            v = fmaxf(v, __shfl_xor(v, 4, 32));
            if (lane == 0)  resp[rowbase + i]     = v;  // pixel M=i
            if (lane == 16) resp[rowbase + 8 + i] = v;  // pixel M=8+i
        }
    }
}

// ---------------- Kernel B: 11x11 NMS peak detect + candidate compaction -----
__global__ __launch_bounds__(256)
void peaksB_kernel(const float* __restrict__ resp, unsigned* __restrict__ counter,
                   uint2* __restrict__ cand) {
    __shared__ float t[26][27];
    const int bx = blockIdx.x, by = blockIdx.y, b = blockIdx.z;
    const int tx = threadIdx.x, ty = threadIdx.y;
    const int tid = ty * 16 + tx;
    const long base = (long)b * HW;

    for (int e = tid; e < 26 * 26; e += 256) {
        int r = e / 26, c = e % 26;
        int gy = by * 16 + r - 5;
        int gx = bx * 16 + c - 5;
        float v = NEGF;
        if (gy >= 0 && gy < HH && gx >= 0 && gx < WW)
            v = fmaxf(resp[base + (long)gy * WW + gx], 0.0f);  // relu
        t[r][c] = v;
    }
    __syncthreads();

    const float r0 = t[ty + 5][tx + 5];
    float pooled = NEGF;
    #pragma unroll
    for (int dy = 0; dy < 11; ++dy)
        #pragma unroll
        for (int dx = 0; dx < 11; ++dx)
            pooled = fmaxf(pooled, t[ty + dy][tx + dx]);

    const int x = bx * 16 + tx;
    const int y = by * 16 + ty;
    if (r0 > 0.5f && r0 == pooled &&
        x >= 5 && x < WW - 5 && y >= 5 && y < HH - 5) {
        unsigned i = atomicAdd(counter, 1u);
        if (i < CAP) {
            uint2 e;
            e.x = __float_as_uint(r0);
            e.y = (unsigned)(base + (long)y * WW + x);
            cand[i] = e;
        }
    }
}

// ---------------- Kernel C: exact global top-4096 (radix select + bitonic) ---
__device__ __forceinline__ bool cmp_before(uint2 a, uint2 b) {
    return (a.x > b.x) || (a.x == b.x && a.y < b.y);  // score desc, idx asc
}

__global__ __launch_bounds__(1024)
void topkC_kernel(const uint2* __restrict__ cand, const unsigned* __restrict__ counter,
                  const float* __restrict__ mask, float* __restrict__ res) {
    __shared__ uint2 top[TOPK];           // 32 KB
    __shared__ unsigned hist[256];
    __shared__ unsigned sh_prefix, sh_k, sh_cnt[2];

    const int tid = threadIdx.x;
    const unsigned n = min(*counter, (unsigned)CAP);

    unsigned thr = 0;
    if (n > TOPK) {
        if (tid == 0) { sh_prefix = 0; sh_k = TOPK; }
        __syncthreads();
        for (int shift = 24; shift >= 0; shift -= 8) {
            for (int i = tid; i < 256; i += 1024) hist[i] = 0;
            __syncthreads();
            const unsigned prefix = sh_prefix;
            const unsigned k = sh_k;
            const unsigned hm = (shift == 24) ? 0u : (0xFFFFFFFFu << (shift + 8));
            for (unsigned i = tid; i < n; i += 1024) {
                unsigned s = cand[i].x;
                if ((s & hm) == prefix) atomicAdd(&hist[(s >> shift) & 255], 1u);
            }
            __syncthreads();
            if (tid == 0) {
                unsigned cum = 0; int bsel = 0;
                for (int bin = 255; bin >= 0; --bin) {
                    unsigned h = hist[bin];
                    if (cum + h >= k) { bsel = bin; break; }
                    cum += h;
                }
                sh_prefix = prefix | ((unsigned)bsel << shift);
                sh_k = k - cum;
            }
            __syncthreads();
        }
        thr = sh_prefix;
    }

    if (tid == 0) { sh_cnt[0] = 0; sh_cnt[1] = 0; }
    __syncthreads();

    unsigned total;
    if (n <= TOPK) {
        for (unsigned i = tid; i < n; i += 1024) {
            unsigned p = atomicAdd(&sh_cnt[0], 1u);
            top[p] = cand[i];
        }
        total = n;
    } else {
        for (unsigned i = tid; i < n; i += 1024) {
            if (cand[i].x > thr) {
                unsigned p = atomicAdd(&sh_cnt[0], 1u);
                top[p] = cand[i];
            }
        }
        __syncthreads();
        const unsigned gt = sh_cnt[0];
        for (unsigned i = tid; i < n; i += 1024) {
            if (cand[i].x == thr) {
                unsigned p = atomicAdd(&sh_cnt[1], 1u);
                if (gt + p < TOPK) top[gt + p] = cand[i];
            }
        }
        total = TOPK;
    }
    __syncthreads();
    for (unsigned i = total + tid; i < TOPK; i += 1024) {
        uint2 s; s.x = 0u; s.y = 0xFFFFFFFFu;
        top[i] = s;
    }
    __syncthreads();

    // bitonic sort, 4096 elems, descending by (score, -idx)
    for (int k = 2; k <= TOPK; k <<= 1) {
        for (int j = k >> 1; j > 0; j >>= 1) {
            for (int i = tid; i < TOPK; i += 1024) {
                int ixj = i ^ j;
                if (ixj > i) {
                    uint2 a = top[i], b = top[ixj];
                    bool desc = ((i & k) == 0);
                    bool swp = desc ? cmp_before(b, a) : cmp_before(a, b);
                    if (swp) { top[i] = b; top[ixj] = a; }
                }
            }
            __syncthreads();
        }
    }

    for (int slot = tid; slot < TOPK; slot += 1024) {
        uint2 e = top[slot];
        float r0 = 0.f, r1 = 0.f, r2 = 0.f, r3 = 0.f, r4 = 0.f;
        if (e.y != 0xFFFFFFFFu) {
            unsigned idx = e.y;
            unsigned bb = idx / HW;
            unsigned rem = idx % HW;
            unsigned y = rem / WW;
            unsigned x = rem % WW;
            r0 = (float)bb; r1 = (float)x; r2 = (float)y;
            r3 = mask[idx];
            r4 = __uint_as_float(e.x);
        }
        res[slot * 5 + 0] = r0;
        res[slot * 5 + 1] = r1;
        res[slot * 5 + 2] = r2;
        res[slot * 5 + 3] = r3;
        res[slot * 5 + 4] = r4;
    }
}

// ---------------- launcher ---------------------------------------------------
extern "C" void kernel_launch(void* const* d_in, const int* in_sizes, int n_in,
                              void* d_out, int out_size, void* d_ws, size_t ws_size,
                              hipStream_t stream) {
    const float* C    = (const float*)d_in[0];
    const float* S    = (const float*)d_in[1];
    // d_in[2] = R, d_in[3] = M : unused by the reference
    const float* mask = (const float*)d_in[4];

    float* res  = (float*)d_out;                 // [4096,5]
    float* resp = (float*)d_out + TOPK * 5;      // conv_resp_out [16,1,768,768]

    unsigned* counter = (unsigned*)d_ws;
    uint2* cand = (uint2*)((char*)d_ws + 16);    // CAP * 8 bytes

    hipMemsetAsync(d_ws, 0, 16, stream);

    convA_kernel<<<dim3(WW / TX, HH / TY, BATCH), 256, 0, stream>>>(C, S, resp);
    peaksB_kernel<<<dim3(WW / 16, HH / 16, BATCH), dim3(16, 16), 0, stream>>>(resp, counter, cand);
    topkC_kernel<<<1, 1024, 0, stream>>>(cand, counter, mask, res);
}